// JointGraphAER_29738353557697
// MI455X (gfx1250) — compile-verified
//
#include <hip/hip_runtime.h>
#include <hip/hip_bf16.h>
#include <math.h>

typedef __attribute__((ext_vector_type(2))) float v2f;
typedef __attribute__((ext_vector_type(8))) float v8f;

#define CDIV(a, b) (((a) + (b) - 1) / (b))
#define LDS_AS __attribute__((address_space(3)))

// ---------------------------------------------------------------------------
// WMMA helper: D = A(16x4) * B(4x16) + C, full fp32 (matches reference math).
// ---------------------------------------------------------------------------
static __device__ __forceinline__ v8f wmma4f32(v2f a, v2f b, v8f c) {
  return __builtin_amdgcn_wmma_f32_16x16x4_f32(false, a, false, b, (short)0, c,
                                               false, false);
}

// ---------------------------------------------------------------------------
// CDNA5 async global->LDS copy (16B per lane), tracked by ASYNCcnt.
// VDST register holds the raw LDS byte offset; VADDR holds the 64-bit global
// address (GV mode). Followed by s_wait_asynccnt + barrier before LDS reads.
// ---------------------------------------------------------------------------
static __device__ __forceinline__ void async_copy_b128(const float* gsrc,
                                                       float* ldst) {
  unsigned loff = (unsigned)(unsigned long long)(LDS_AS float*)ldst;
  unsigned long long ga = (unsigned long long)gsrc;
  asm volatile("global_load_async_to_lds_b128 %0, %1, off"
               :
               : "v"(loff), "v"(ga)
               : "memory");
}

static __device__ __forceinline__ void wait_async0() {
  asm volatile("s_wait_asynccnt 0x0" ::: "memory");
}

// ---------------------------------------------------------------------------
// Zero-fill
// ---------------------------------------------------------------------------
__global__ void fill_zero(float* __restrict__ p, size_t n) {
  size_t i = (size_t)blockIdx.x * 256 + threadIdx.x;
  if (i < n) p[i] = 0.0f;
}

// ---------------------------------------------------------------------------
// Seed agg with h (stride IND -> stride 64, zero-pad cols >= IND).
// After the message pass, agg[n] = h[n] + sum_j relu(...) : exactly the MLP
// input, making the MLP tile staging a pure copy (async-LDS friendly).
// ---------------------------------------------------------------------------
template <int IND>
__global__ void seed_agg(const float* __restrict__ h, float* __restrict__ agg,
                         int N) {
  int idx = blockIdx.x * 256 + threadIdx.x;
  if (idx >= N * 64) return;
  int n = idx >> 6, k = idx & 63;
  agg[idx] = (k < IND) ? h[(size_t)n * IND + k] : 0.0f;
}

// ---------------------------------------------------------------------------
// GINE message: per edge e, dim j:
//   msg = relu(h[src][j] + (W_e @ ea_e)[j] + b_e[j]);  atomicAdd(agg[dst][j])
// ---------------------------------------------------------------------------
template <int IND>
__global__ __launch_bounds__(256) void gine_message(
    const float* __restrict__ h, const float* __restrict__ ea,
    const int* __restrict__ ei, const float* __restrict__ We,
    const float* __restrict__ be, float* __restrict__ agg, int E) {
  __shared__ float sW[IND * 17];  // odd pad -> conflict-free on 64 banks
  __shared__ float sB[IND];
  const int tid = threadIdx.x;
  for (int idx = tid; idx < IND * 16; idx += 256) {
    int j = idx >> 4, k = idx & 15;
    sW[j * 17 + k] = We[idx];
  }
  if (tid < IND) sB[tid] = be[tid];
  __syncthreads();

  constexpr int EPB = 256 / IND;
  const int e = blockIdx.x * EPB + tid / IND;
  const int j = tid % IND;
  if (e >= E) return;
  const int src = ei[e];
  const int dst = ei[E + e];
  const float* eap = ea + (size_t)e * 16;
  float acc = sB[j];
#pragma unroll
  for (int k = 0; k < 16; ++k) acc += sW[j * 17 + k] * eap[k];
  float msg = fmaxf(h[(size_t)src * IND + j] + acc, 0.0f);
  atomicAdd(&agg[(size_t)dst * 64 + j], msg);
}

// ---------------------------------------------------------------------------
// Fused node MLP (WMMA f32 16x16x4), async-LDS staged:
//   out = [BN+ReLU]( W2 @ relu(W1 @ in + b1) + b2 )
// in has stride 64 (agg buffer or h_nodes). 64 nodes/block, 4 waves.
// LDS row stride = dim+4 -> 16B-aligned chunks AND conflict-free b64 reads.
// ---------------------------------------------------------------------------
template <int IND, int OUT2, bool USE_BN>
__global__ __launch_bounds__(128) void node_mlp_wmma(
    const float* __restrict__ in, const float* __restrict__ W1,
    const float* __restrict__ b1, const float* __restrict__ W2,
    const float* __restrict__ b2, const float* __restrict__ bng,
    const float* __restrict__ bnb, const float* __restrict__ bnm,
    const float* __restrict__ bnv, float* __restrict__ hout, int nNodes) {
  constexpr int MID = 64;
  constexpr int P1 = IND + 4;
  constexpr int PM = MID + 4;
  __shared__ float sIn[64 * P1];
  __shared__ float sMid[64 * PM];
  __shared__ float sW1[MID * P1];
  __shared__ float sW2[OUT2 * PM];

  const int tid = threadIdx.x;
  const int wave = tid >> 5;
  const int lane = tid & 31;
  const int lo = lane & 15, hi = lane >> 4;
  const int nodeBase = blockIdx.x * 64;

  // --- async global->LDS staging (pure copies, 16B chunks) ---
  constexpr int C1 = IND / 4;  // chunks per input/W1 row
  for (int idx = tid; idx < 64 * C1; idx += 128) {
    int r = idx / C1, c = idx - r * C1;
    int gn = nodeBase + r;
    if (gn < nNodes)  // OOB rows stay garbage; their outputs are never stored
      async_copy_b128(&in[(size_t)gn * 64 + c * 4], &sIn[r * P1 + c * 4]);
  }
  for (int idx = tid; idx < MID * C1; idx += 128) {
    int r = idx / C1, c = idx - r * C1;
    async_copy_b128(&W1[(size_t)r * IND + c * 4], &sW1[r * P1 + c * 4]);
  }
  constexpr int C2 = MID / 4;  // chunks per W2 row
  for (int idx = tid; idx < OUT2 * C2; idx += 128) {
    int r = idx / C2, c = idx - r * C2;
    async_copy_b128(&W2[(size_t)r * MID + c * 4], &sW2[r * PM + c * 4]);
  }
  wait_async0();
  __syncthreads();

  // --- GEMM1: mid = relu(in @ W1^T + b1) ---
  for (int nt = 0; nt < MID / 16; ++nt) {
    v8f acc;
    float bv = b1[nt * 16 + lo];
#pragma unroll
    for (int i = 0; i < 8; ++i) acc[i] = bv;
#pragma unroll
    for (int k0 = 0; k0 < IND; k0 += 4) {
      v2f a = *(const v2f*)&sIn[(wave * 16 + lo) * P1 + k0 + 2 * hi];
      v2f b = *(const v2f*)&sW1[(nt * 16 + lo) * P1 + k0 + 2 * hi];
      acc = wmma4f32(a, b, acc);
    }
#pragma unroll
    for (int v = 0; v < 8; ++v) {
      sMid[(wave * 16 + v + 8 * hi) * PM + nt * 16 + lo] = fmaxf(acc[v], 0.0f);
    }
  }
  __syncthreads();

  // --- GEMM2: out = epilogue(mid @ W2^T + b2) ---
  for (int nt = 0; nt < OUT2 / 16; ++nt) {
    const int o = nt * 16 + lo;
    v8f acc;
    float bv = b2[o];
#pragma unroll
    for (int i = 0; i < 8; ++i) acc[i] = bv;
#pragma unroll
    for (int k0 = 0; k0 < MID; k0 += 4) {
      v2f a = *(const v2f*)&sMid[(wave * 16 + lo) * PM + k0 + 2 * hi];
      v2f b = *(const v2f*)&sW2[(nt * 16 + lo) * PM + k0 + 2 * hi];
      acc = wmma4f32(a, b, acc);
    }
    float s = 1.0f, sh = 0.0f;
    if (USE_BN) {
      float g = bng[o], bb = bnb[o], m = bnm[o], vv = bnv[o];
      s = g * rsqrtf(vv + 1e-5f);
      sh = bb - m * s;
    }
#pragma unroll
    for (int v = 0; v < 8; ++v) {
      int gn = nodeBase + wave * 16 + v + 8 * hi;
      if (gn < nNodes) {
        float x = acc[v];
        if (USE_BN) x = fmaxf(x * s + sh, 0.0f);
        hout[(size_t)gn * OUT2 + o] = x;
      }
    }
  }
}

// ---------------------------------------------------------------------------
// Mean pool (atomics; z-sums are L2-resident)
// ---------------------------------------------------------------------------
__global__ void pool_accum(const float* __restrict__ h,
                           const int* __restrict__ batch,
                           float* __restrict__ zsum, float* __restrict__ cnt,
                           int N) {
  int idx = blockIdx.x * 256 + threadIdx.x;
  if (idx >= N * 64) return;
  int n = idx >> 6, c = idx & 63;
  int g = batch[n];
  atomicAdd(&zsum[g * 64 + c], h[idx]);
  if (c == 0) atomicAdd(&cnt[g], 1.0f);
}

__global__ void pool_finalize(const float* __restrict__ zsum,
                              const float* __restrict__ cnt,
                              const float* __restrict__ numf,
                              float* __restrict__ zgraph, int G) {
  int idx = blockIdx.x * 256 + threadIdx.x;
  if (idx >= G * 68) return;
  int g = idx / 68, c = idx - g * 68;
  zgraph[idx] = (c < 64) ? zsum[g * 64 + c] / fmaxf(cnt[g], 1.0f)
                         : numf[g * 4 + (c - 64)];
}

// ---------------------------------------------------------------------------
// Generic small GEMM (rows x K) @ W[OUT x K]^T + bias, one wave per 16x16 tile.
// mode: 0 = bias only, 1 = ReLU, 2 = BN+ReLU, 3 = sigmoid
// ---------------------------------------------------------------------------
__global__ __launch_bounds__(32) void gemm_wmma_ep(
    const float* __restrict__ A, const float* __restrict__ W,
    const float* __restrict__ bias, const float* __restrict__ bng,
    const float* __restrict__ bnb, const float* __restrict__ bnm,
    const float* __restrict__ bnv, float* __restrict__ out, int K, int OUT,
    int mode) {
  const int lane = threadIdx.x;
  const int lo = lane & 15, hi = lane >> 4;
  const int row = blockIdx.x * 16 + lo;
  const int col = blockIdx.y * 16 + lo;
  const float* arow = A + (size_t)row * K;
  const float* wrow = W + (size_t)col * K;
  v8f acc;
  float bv = bias[col];
#pragma unroll
  for (int i = 0; i < 8; ++i) acc[i] = bv;
  for (int k0 = 0; k0 < K; k0 += 4) {
    v2f a, b;
    a.x = arow[k0 + 2 * hi];
    a.y = arow[k0 + 2 * hi + 1];
    b.x = wrow[k0 + 2 * hi];
    b.y = wrow[k0 + 2 * hi + 1];
    acc = wmma4f32(a, b, acc);
  }
  float s = 1.0f, sh = 0.0f;
  if (mode == 2) {
    float g = bng[col], bb = bnb[col], m = bnm[col], vv = bnv[col];
    s = g * rsqrtf(vv + 1e-5f);
    sh = bb - m * s;
  }
#pragma unroll
  for (int v = 0; v < 8; ++v) {
    int r = blockIdx.x * 16 + v + 8 * hi;
    float x = acc[v];
    if (mode == 2)
      x = fmaxf(x * s + sh, 0.0f);
    else if (mode == 1)
      x = fmaxf(x, 0.0f);
    else if (mode == 3)
      x = 1.0f / (1.0f + __expf(-x));
    out[(size_t)r * OUT + blockIdx.y * 16 + lo] = x;
  }
}

__global__ void ewise_mul(const float* __restrict__ a,
                          const float* __restrict__ b, float* __restrict__ o,
                          int n) {
  int i = blockIdx.x * 256 + threadIdx.x;
  if (i < n) o[i] = a[i] * b[i];
}

// head2: y[g] = hh[g,:] . w + b   (OUT=1), one wave per graph
__global__ __launch_bounds__(32) void head2_dot(const float* __restrict__ hh,
                                                const float* __restrict__ w,
                                                const float* __restrict__ b,
                                                float* __restrict__ y) {
  int g = blockIdx.x;
  int lane = threadIdx.x;
  float p = 0.0f;
  for (int k = lane; k < 256; k += 32) p += hh[(size_t)g * 256 + k] * w[k];
#pragma unroll
  for (int off = 16; off > 0; off >>= 1) p += __shfl_xor(p, off, 32);
  if (lane == 0) y[g] = p + b[0];
}

// ---------------------------------------------------------------------------
// Launch
// ---------------------------------------------------------------------------
extern "C" void kernel_launch(void* const* d_in, const int* in_sizes, int n_in,
                              void* d_out, int out_size, void* d_ws,
                              size_t ws_size, hipStream_t stream) {
  (void)n_in;
  (void)out_size;
  (void)ws_size;
  const float* x = (const float*)d_in[0];
  const int* ei = (const int*)d_in[1];
  const float* ea = (const float*)d_in[2];
  const int* batch = (const int*)d_in[3];
  const float* numf = (const float*)d_in[4];
  const int N = in_sizes[3];
  const int E = in_sizes[1] / 2;
  const int G = in_sizes[4] / 4;
  auto P = [&](int i) { return (const float*)d_in[i]; };

  // param indices (setup_inputs insertion order)
  const int L0 = 5, L1 = 15, L2 = 25;      // edge.w,b mlp1.w,b mlp2.w,b bn g,b,m,v
  const int DEC = 35;                      // l1.w, l1.b, l2.w, l2.b
  const int RB0 = 39, RB1 = 45, RB2 = 51;  // lin.w, lin.b, bn g,b,m,v
  const int SE1 = 57, SE2 = 59, HD1 = 61, HBN = 63, HD2 = 67;

  // workspace
  float* ws = (float*)d_ws;
  size_t o = 0;
  float* agg = ws + o;  o += (size_t)N * 64;
  float* hA = ws + o;   o += (size_t)N * 64;
  float* hB = ws + o;   o += (size_t)N * 64;
  float* zsum = ws + o; o += (size_t)G * 64;
  float* cnt = ws + o;  o += (size_t)G;
  float* g1 = ws + o;   o += (size_t)G * 512;
  float* g2 = ws + o;   o += (size_t)G * 512;
  float* g3 = ws + o;   o += (size_t)G * 512;
  float* seh = ws + o;  o += (size_t)G * 64;
  float* sew = ws + o;  o += (size_t)G * 512;
  float* gse = ws + o;  o += (size_t)G * 512;
  float* hh = ws + o;   o += (size_t)G * 256;

  // output regions: y_hat | x_logits | h_nodes | z_graph
  float* out_y = (float*)d_out;
  float* out_xl = out_y + G;
  float* out_h = out_xl + (size_t)N * 32;
  float* out_zg = out_h + (size_t)N * 64;

  const int nAggGrid = CDIV(N * 64, 256);

  // ---- encoder layer 0 (IND=32, in = x) -> hA
  seed_agg<32><<<nAggGrid, 256, 0, stream>>>(x, agg, N);
  gine_message<32><<<CDIV(E, 8), 256, 0, stream>>>(x, ea, ei, P(L0), P(L0 + 1),
                                                   agg, E);
  node_mlp_wmma<32, 64, true><<<CDIV(N, 64), 128, 0, stream>>>(
      agg, P(L0 + 2), P(L0 + 3), P(L0 + 4), P(L0 + 5), P(L0 + 6), P(L0 + 7),
      P(L0 + 8), P(L0 + 9), hA, N);

  // ---- encoder layer 1 (IND=64, in = hA) -> hB
  seed_agg<64><<<nAggGrid, 256, 0, stream>>>(hA, agg, N);
  gine_message<64><<<CDIV(E, 4), 256, 0, stream>>>(hA, ea, ei, P(L1),
                                                   P(L1 + 1), agg, E);
  node_mlp_wmma<64, 64, true><<<CDIV(N, 64), 128, 0, stream>>>(
      agg, P(L1 + 2), P(L1 + 3), P(L1 + 4), P(L1 + 5), P(L1 + 6), P(L1 + 7),
      P(L1 + 8), P(L1 + 9), hB, N);

  // ---- encoder layer 2 (IND=64, in = hB) -> h_nodes output region
  seed_agg<64><<<nAggGrid, 256, 0, stream>>>(hB, agg, N);
  gine_message<64><<<CDIV(E, 4), 256, 0, stream>>>(hB, ea, ei, P(L2),
                                                   P(L2 + 1), agg, E);
  node_mlp_wmma<64, 64, true><<<CDIV(N, 64), 128, 0, stream>>>(
      agg, P(L2 + 2), P(L2 + 3), P(L2 + 4), P(L2 + 5), P(L2 + 6), P(L2 + 7),
      P(L2 + 8), P(L2 + 9), out_h, N);

  // ---- decoder: x_logits = l2(relu(l1(h_nodes)))  (no BN, no final ReLU)
  node_mlp_wmma<64, 32, false><<<CDIV(N, 64), 128, 0, stream>>>(
      out_h, P(DEC), P(DEC + 1), P(DEC + 2), P(DEC + 3), nullptr, nullptr,
      nullptr, nullptr, out_xl, N);

  // ---- mean pool + concat extras -> z_graph
  fill_zero<<<CDIV((size_t)G * 64 + G, 256), 256, 0, stream>>>(
      zsum, (size_t)G * 64 + G);  // zsum and cnt are adjacent
  pool_accum<<<CDIV((size_t)N * 64, 256), 256, 0, stream>>>(out_h, batch, zsum,
                                                            cnt, N);
  pool_finalize<<<CDIV(G * 68, 256), 256, 0, stream>>>(zsum, cnt, numf, out_zg,
                                                       G);

  // ---- regressor
  dim3 w32(32, 1, 1);
  gemm_wmma_ep<<<dim3(G / 16, 32), w32, 0, stream>>>(
      out_zg, P(RB0), P(RB0 + 1), P(RB0 + 2), P(RB0 + 3), P(RB0 + 4),
      P(RB0 + 5), g1, 68, 512, 2);
  gemm_wmma_ep<<<dim3(G / 16, 32), w32, 0, stream>>>(
      g1, P(RB1), P(RB1 + 1), P(RB1 + 2), P(RB1 + 3), P(RB1 + 4), P(RB1 + 5),
      g2, 512, 512, 2);
  gemm_wmma_ep<<<dim3(G / 16, 32), w32, 0, stream>>>(
      g2, P(RB2), P(RB2 + 1), P(RB2 + 2), P(RB2 + 3), P(RB2 + 4), P(RB2 + 5),
      g3, 512, 512, 2);
  // SE block
  gemm_wmma_ep<<<dim3(G / 16, 4), w32, 0, stream>>>(
      g3, P(SE1), P(SE1 + 1), nullptr, nullptr, nullptr, nullptr, seh, 512, 64,
      1);
  gemm_wmma_ep<<<dim3(G / 16, 32), w32, 0, stream>>>(
      seh, P(SE2), P(SE2 + 1), nullptr, nullptr, nullptr, nullptr, sew, 64, 512,
      3);
  ewise_mul<<<CDIV(G * 512, 256), 256, 0, stream>>>(g3, sew, gse, G * 512);
  // head
  gemm_wmma_ep<<<dim3(G / 16, 16), w32, 0, stream>>>(
      gse, P(HD1), P(HD1 + 1), P(HBN), P(HBN + 1), P(HBN + 2), P(HBN + 3), hh,
      512, 256, 2);
  head2_dot<<<G, 32, 0, stream>>>(hh, P(HD2), P(HD2 + 1), out_y);
}